// GFM_4870492913893
// MI455X (gfx1250) — compile-verified
//
#include <hip/hip_runtime.h>
#include <math.h>

typedef float v2f __attribute__((ext_vector_type(2)));
typedef float v8f __attribute__((ext_vector_type(8)));

#define SENT 0x7FFFFFFF
#define DIM 64

// ---------------------------------------------------------------------------
// Kernel 1: init flag[] to SENT, zero compact accumulators.
// ---------------------------------------------------------------------------
__global__ void gfm_init_kernel(int* __restrict__ flag, float* __restrict__ s_comp,
                                float* __restrict__ sq_comp, int n_ent, int batch) {
    int idx    = blockIdx.x * blockDim.x + threadIdx.x;
    int stride = gridDim.x * blockDim.x;
    for (int j = idx; j < n_ent; j += stride) flag[j] = SENT;
    int tot = batch * DIM;
    for (int j = idx; j < tot; j += stride) { s_comp[j] = 0.0f; sq_comp[j] = 0.0f; }
}

// ---------------------------------------------------------------------------
// Kernel 2: flag batch entities with their smallest batch index (deterministic).
// ---------------------------------------------------------------------------
__global__ void gfm_flag_kernel(const int* __restrict__ ii, int* __restrict__ flag,
                                int batch) {
    int b = blockIdx.x * blockDim.x + threadIdx.x;
    if (b < batch) atomicMin(&flag[ii[b]], b);
}

// ---------------------------------------------------------------------------
// Kernel 3: edge scan. One thread per edge tests flag[dst]; contributing edges
// (~1.6%) are processed cooperatively by the whole wave32: coalesced 64-float
// row gather, on-the-fly max_norm=1 rescale, atomic accumulate s and s^2 into
// compact per-slot rows.
// ---------------------------------------------------------------------------
__global__ void gfm_edge_kernel(const float* __restrict__ ent,
                                const int* __restrict__ esrc,
                                const int* __restrict__ edst,
                                const int* __restrict__ flag,
                                float* __restrict__ s_comp,
                                float* __restrict__ sq_comp, int n_edges) {
    int e    = blockIdx.x * blockDim.x + threadIdx.x;
    int lane = threadIdx.x & 31;

    int  slot = SENT;
    int  src  = 0;
    bool act  = false;
    if (e < n_edges) {
        int d = edst[e];
        slot  = flag[d];
        if (slot != SENT) { src = esrc[e]; act = true; }
    }

    unsigned long long m = __ballot(act);   // uniform across the wave
    while (m) {
        int bit = __ffsll((long long)m) - 1;
        m &= m - 1;
        int t_slot = __shfl(slot, bit, 32);
        int t_src  = __shfl(src,  bit, 32);

        const float* row = ent + (size_t)t_src * DIM;
        float v0 = row[lane];
        float v1 = row[lane + 32];
        float ss = v0 * v0 + v1 * v1;
        #pragma unroll
        for (int off = 16; off; off >>= 1) ss += __shfl_xor(ss, off, 32);
        float sc = fminf(1.0f, 1.0f / fmaxf(sqrtf(ss), 1e-12f));
        float h0 = v0 * sc, h1 = v1 * sc;

        float* sp = s_comp  + (size_t)t_slot * DIM;
        float* qp = sq_comp + (size_t)t_slot * DIM;
        atomicAdd(sp + lane,      h0);
        atomicAdd(sp + lane + 32, h1);
        atomicAdd(qp + lane,      h0 * h0);
        atomicAdd(qp + lane + 32, h1 * h1);
    }
}

// ---------------------------------------------------------------------------
// Kernel 4: per 16-batch tile, stage renormed users and FM items into LDS,
// then compute the 16 dot products as the diagonal of a 16x16 GEMM using
// V_WMMA_F32_16X16X4_F32 chained over K=64 (16 steps of k=4), f32 throughout.
// One wave32 per tile; EXEC all-ones at the WMMA (no divergence).
// ---------------------------------------------------------------------------
__global__ __launch_bounds__(32) void gfm_fm_wmma_kernel(
        const float* __restrict__ user_emb, const float* __restrict__ ent,
        const int* __restrict__ uu, const int* __restrict__ ii,
        const int* __restrict__ flag, const float* __restrict__ s_comp,
        const float* __restrict__ sq_comp, float* __restrict__ out, int batch) {
    __shared__ float ldsU[16 * DIM];
    __shared__ float ldsI[16 * DIM];

    int b0   = blockIdx.x * 16;
    int lane = threadIdx.x;   // 0..31

    // ---- phase 1: stage 16 user rows (renormed) + 16 item rows (s*s-sq+x) ----
    for (int r = 0; r < 16; ++r) {
        int b = b0 + r;
        // user row, max_norm=1
        const float* ur = user_emb + (size_t)uu[b] * DIM;
        float w0 = ur[lane], w1 = ur[lane + 32];
        float ssu = w0 * w0 + w1 * w1;
        #pragma unroll
        for (int off = 16; off; off >>= 1) ssu += __shfl_xor(ssu, off, 32);
        float scu = fminf(1.0f, 1.0f / fmaxf(sqrtf(ssu), 1e-12f));
        ldsU[r * DIM + lane]      = w0 * scu;
        ldsU[r * DIM + lane + 32] = w1 * scu;

        // item row: x_new = s*s - sq + renorm(x)
        int eidx = ii[b];
        const float* xr = ent + (size_t)eidx * DIM;
        float x0 = xr[lane], x1 = xr[lane + 32];
        float ssx = x0 * x0 + x1 * x1;
        #pragma unroll
        for (int off = 16; off; off >>= 1) ssx += __shfl_xor(ssx, off, 32);
        float scx = fminf(1.0f, 1.0f / fmaxf(sqrtf(ssx), 1e-12f));
        x0 *= scx; x1 *= scx;

        int slot = flag[eidx];  // always valid: every i[b] was flagged
        const float* sp = s_comp  + (size_t)slot * DIM;
        const float* qp = sq_comp + (size_t)slot * DIM;
        float s0 = sp[lane], s1 = sp[lane + 32];
        float q0 = qp[lane], q1 = qp[lane + 32];
        ldsI[r * DIM + lane]      = s0 * s0 - q0 + x0;
        ldsI[r * DIM + lane + 32] = s1 * s1 - q1 + x1;
    }
    __syncthreads();

    // ---- phase 2: D = U(16x64) x I^T(64x16) via 16 chained WMMA f32 16x16x4 ----
    // A(16x4): lane L<16 -> {A[L][k],A[L][k+1]}, lane L+16 -> {A[L][k+2],A[L][k+3]}
    // B(4x16): VGPR0 -> K in {k,k+2}, VGPR1 -> K in {k+1,k+3}; B[K][N] = items[N][K]
    // => both fragments index identically: row = lane&15, khalf = 2*(lane>>4)
    int row   = lane & 15;
    int khalf = (lane >> 4) << 1;
    v8f c = {};
    #pragma unroll
    for (int ko = 0; ko < 16; ++ko) {
        int k = ko * 4 + khalf;
        v2f a, bb;
        a.x  = ldsU[row * DIM + k];
        a.y  = ldsU[row * DIM + k + 1];
        bb.x = ldsI[row * DIM + k];
        bb.y = ldsI[row * DIM + k + 1];
        c = __builtin_amdgcn_wmma_f32_16x16x4_f32(
                /*neg_a=*/false, a, /*neg_b=*/false, bb,
                /*c_mod=*/(short)0, c, /*reuse_a=*/false, /*reuse_b=*/false);
    }

    // ---- diagonal extraction: (m,m) for m<8 -> lane m, c[m];
    //      m in 8..15 -> lane m+16 (24..31), c[lane-24] ----
    float dv = 0.0f;
    int   ob = -1;
    if (lane < 8)        { dv = c[lane];      ob = b0 + lane; }
    else if (lane >= 24) { dv = c[lane - 24]; ob = b0 + 8 + (lane - 24); }
    if (ob >= 0 && ob < batch) out[ob] = 1.0f / (1.0f + expf(-dv));
}

// ---------------------------------------------------------------------------
// Launch. Inputs (setup_inputs order):
//   0: user_emb f32 [N_USERS*64]   1: entity_emb f32 [N_ENT*64]
//   2: u i32 [B]   3: i i32 [B]   4: edge_src i32 [E]   5: edge_dst i32 [E]
// out: f32 [B]. ws: flag(4MB) | s_comp(4MB) | sq_comp(4MB).
// ---------------------------------------------------------------------------
extern "C" void kernel_launch(void* const* d_in, const int* in_sizes, int n_in,
                              void* d_out, int out_size, void* d_ws, size_t ws_size,
                              hipStream_t stream) {
    const float* user_emb = (const float*)d_in[0];
    const float* ent      = (const float*)d_in[1];
    const int*   uu       = (const int*)d_in[2];
    const int*   ii       = (const int*)d_in[3];
    const int*   esrc     = (const int*)d_in[4];
    const int*   edst     = (const int*)d_in[5];
    float*       out      = (float*)d_out;

    const int n_ent   = in_sizes[1] / DIM;
    const int batch   = in_sizes[2];
    const int n_edges = in_sizes[4];

    char* ws = (char*)d_ws;
    size_t off = 0;
    int* flag = (int*)(ws + off);
    off += ((size_t)n_ent * sizeof(int) + 255) & ~(size_t)255;
    float* s_comp = (float*)(ws + off);
    off += ((size_t)batch * DIM * sizeof(float) + 255) & ~(size_t)255;
    float* sq_comp = (float*)(ws + off);

    gfm_init_kernel<<<4096, 256, 0, stream>>>(flag, s_comp, sq_comp, n_ent, batch);
    gfm_flag_kernel<<<(batch + 255) / 256, 256, 0, stream>>>(ii, flag, batch);
    gfm_edge_kernel<<<(n_edges + 255) / 256, 256, 0, stream>>>(
        ent, esrc, edst, flag, s_comp, sq_comp, n_edges);
    gfm_fm_wmma_kernel<<<(batch + 15) / 16, 32, 0, stream>>>(
        user_emb, ent, uu, ii, flag, s_comp, sq_comp, out, batch);
}